// DeeperGCN_31799937860080
// MI455X (gfx1250) — compile-verified
//
#include <hip/hip_runtime.h>
#include <hip/hip_bf16.h>

#define N_NODES 100000
#define N_EDGES 1200000
#define HID 64
#define H2 128
#define OUTD 112
#define LAYERS 7
#define EPS_MSG 1e-7f
#define LN_EPS 1e-5f

typedef float v2f __attribute__((ext_vector_type(2)));
typedef float v8f __attribute__((ext_vector_type(8)));

// ---------------------------------------------------------------------------
// Generic fp32 WMMA GEMM: C[M,N] = A[M,K] @ B[K,N] + bias, one 16-row tile
// per wave. Uses V_WMMA_F32_16X16X4_F32 (exact fp32 FMA).
// A-layout (16x4 f32): lane m = lane&15, kb = 2*(lane>=16), a = A[m][k+kb..+1]
// B-layout (4x16 f32): lane n = lane&15, b = B[k+kb..+1][n]
// C-layout: vgpr r, half = lane>=16 -> row r + 8*half, col = lane&15
// ---------------------------------------------------------------------------
template <int K, int NT>
__global__ __launch_bounds__(128) void gemm_bias_wmma(
    const float* __restrict__ A, int lda,
    const float* __restrict__ B, int ldb,
    const float* __restrict__ bias,
    float* __restrict__ C, int ldc, int mtiles) {
  int wid = threadIdx.x >> 5;
  int lane = threadIdx.x & 31;
  int tile = blockIdx.x * 4 + wid;
  if (tile >= mtiles) return;
  int half = lane >> 4;
  int ln16 = lane & 15;
  int kb = half * 2;
  const float* arow = A + (size_t)(tile * 16 + ln16) * lda;

  v8f acc[NT];
#pragma unroll
  for (int nt = 0; nt < NT; ++nt) {
    float bb = bias[nt * 16 + ln16];
    acc[nt] = (v8f){bb, bb, bb, bb, bb, bb, bb, bb};
  }
#pragma unroll
  for (int k = 0; k < K; k += 4) {
    v2f a;
    a[0] = arow[k + kb];
    a[1] = arow[k + kb + 1];
#pragma unroll
    for (int nt = 0; nt < NT; ++nt) {
      int col = nt * 16 + ln16;
      v2f b;
      b[0] = B[(size_t)(k + kb) * ldb + col];
      b[1] = B[(size_t)(k + kb + 1) * ldb + col];
      acc[nt] = __builtin_amdgcn_wmma_f32_16x16x4_f32(
          false, a, false, b, (short)0, acc[nt], false, false);
    }
  }
#pragma unroll
  for (int nt = 0; nt < NT; ++nt) {
    int col = nt * 16 + ln16;
#pragma unroll
    for (int r = 0; r < 8; ++r) {
      C[(size_t)(tile * 16 + r + half * 8) * ldc + col] = acc[nt][r];
    }
  }
}

// ---------------------------------------------------------------------------
// x = relu(LayerNorm(h)); optionally zero mx/s/agg for the upcoming layer.
// One wave per 64-wide row.
// ---------------------------------------------------------------------------
__global__ __launch_bounds__(256) void ln_relu_kernel(
    const float* __restrict__ h, const float* __restrict__ g,
    const float* __restrict__ b, float* __restrict__ x,
    float* __restrict__ mx, float* __restrict__ ssum, float* __restrict__ agg,
    int nrows, int zero_aux) {
  int wid = threadIdx.x >> 5;
  int lane = threadIdx.x & 31;
  int row = blockIdx.x * 8 + wid;
  if (row >= nrows) return;
  size_t base = (size_t)row * HID;
  float v0 = h[base + lane];
  float v1 = h[base + lane + 32];
  float p = v0 + v1;
#pragma unroll
  for (int m = 16; m >= 1; m >>= 1) p += __shfl_xor(p, m, 32);
  float mu = p * (1.f / 64.f);
  float d0 = v0 - mu, d1 = v1 - mu;
  float q = d0 * d0 + d1 * d1;
#pragma unroll
  for (int m = 16; m >= 1; m >>= 1) q += __shfl_xor(q, m, 32);
  float rstd = rsqrtf(q * (1.f / 64.f) + LN_EPS);
  x[base + lane] = fmaxf(d0 * rstd * g[lane] + b[lane], 0.f);
  x[base + lane + 32] = fmaxf(d1 * rstd * g[lane + 32] + b[lane + 32], 0.f);
  if (zero_aux) {
    mx[base + lane] = 0.f;      mx[base + lane + 32] = 0.f;
    ssum[base + lane] = 0.f;    ssum[base + lane + 32] = 0.f;
    agg[base + lane] = 0.f;     agg[base + lane + 32] = 0.f;
  }
}

// ---------------------------------------------------------------------------
// Edge pass 1: m = relu(x[src]+eh)+eps; atomicMax logits into mx[dst].
// 4 threads per edge, 16 features each (float4 x4).
// ---------------------------------------------------------------------------
__global__ __launch_bounds__(256) void edge_pass1_kernel(
    const float* __restrict__ x, const float* __restrict__ eh,
    const int* __restrict__ src, const int* __restrict__ dst,
    const float* __restrict__ betas, int layer, float* __restrict__ mx) {
  int gid = blockIdx.x * 256 + threadIdx.x;
  int e = gid >> 2;
  if (e >= N_EDGES) return;
  int qd = gid & 3;
  float beta = betas[layer];
  int sv = src[e], dv = dst[e];
  const float4* xp = (const float4*)(x + (size_t)sv * HID + qd * 16);
  const float4* ep = (const float4*)(eh + (size_t)e * HID + qd * 16);
  float* mp = mx + (size_t)dv * HID + qd * 16;
#pragma unroll
  for (int j = 0; j < 4; ++j) {
    float4 a = xp[j];
    float4 bb = ep[j];
    float m0 = fmaxf(a.x + bb.x, 0.f) + EPS_MSG;
    float m1 = fmaxf(a.y + bb.y, 0.f) + EPS_MSG;
    float m2 = fmaxf(a.z + bb.z, 0.f) + EPS_MSG;
    float m3 = fmaxf(a.w + bb.w, 0.f) + EPS_MSG;
    atomicMax(mp + j * 4 + 0, m0 * beta);
    atomicMax(mp + j * 4 + 1, m1 * beta);
    atomicMax(mp + j * 4 + 2, m2 * beta);
    atomicMax(mp + j * 4 + 3, m3 * beta);
  }
}

// ---------------------------------------------------------------------------
// Edge pass 2: e = exp(m*beta - mx[dst]); s[dst]+=e; agg[dst]+=m*e.
// (agg/s computed later; shift-invariant so mx >= true max is fine)
// ---------------------------------------------------------------------------
__global__ __launch_bounds__(256) void edge_pass2_kernel(
    const float* __restrict__ x, const float* __restrict__ eh,
    const int* __restrict__ src, const int* __restrict__ dst,
    const float* __restrict__ betas, int layer,
    const float* __restrict__ mx, float* __restrict__ ssum,
    float* __restrict__ agg) {
  int gid = blockIdx.x * 256 + threadIdx.x;
  int e = gid >> 2;
  if (e >= N_EDGES) return;
  int qd = gid & 3;
  float beta = betas[layer];
  int sv = src[e], dv = dst[e];
  const float4* xp = (const float4*)(x + (size_t)sv * HID + qd * 16);
  const float4* ep = (const float4*)(eh + (size_t)e * HID + qd * 16);
  const float* mxp = mx + (size_t)dv * HID + qd * 16;
  float* sp = ssum + (size_t)dv * HID + qd * 16;
  float* ap = agg + (size_t)dv * HID + qd * 16;
#pragma unroll
  for (int j = 0; j < 4; ++j) {
    float4 a = xp[j];
    float4 bb = ep[j];
    float m0 = fmaxf(a.x + bb.x, 0.f) + EPS_MSG;
    float m1 = fmaxf(a.y + bb.y, 0.f) + EPS_MSG;
    float m2 = fmaxf(a.z + bb.z, 0.f) + EPS_MSG;
    float m3 = fmaxf(a.w + bb.w, 0.f) + EPS_MSG;
    float e0 = expf(m0 * beta - mxp[j * 4 + 0]);
    float e1 = expf(m1 * beta - mxp[j * 4 + 1]);
    float e2 = expf(m2 * beta - mxp[j * 4 + 2]);
    float e3 = expf(m3 * beta - mxp[j * 4 + 3]);
    atomicAdd(sp + j * 4 + 0, e0);
    atomicAdd(sp + j * 4 + 1, e1);
    atomicAdd(sp + j * 4 + 2, e2);
    atomicAdd(sp + j * 4 + 3, e3);
    atomicAdd(ap + j * 4 + 0, m0 * e0);
    atomicAdd(ap + j * 4 + 1, m1 * e1);
    atomicAdd(ap + j * 4 + 2, m2 * e2);
    atomicAdd(ap + j * 4 + 3, m3 * e3);
  }
}

// ---------------------------------------------------------------------------
// Fused GENConv MLP per 16-node tile (one wave each, 2 waves/block):
//   y = x + agg/s ; t = relu(LN(y@W1+b1)) ; h += t@W2 + b2
// LayerNorm of GEMM1 output done in registers via half-wave shfl_xor
// reductions; transpose to A-layout via LDS.
// ---------------------------------------------------------------------------
__global__ __launch_bounds__(64) void mlp_fused_wmma(
    const float* __restrict__ x, const float* __restrict__ agg,
    const float* __restrict__ ssum, const float* __restrict__ W1,
    const float* __restrict__ b1, const float* __restrict__ g1,
    const float* __restrict__ bt1, const float* __restrict__ W2,
    const float* __restrict__ b2, float* __restrict__ h) {
  __shared__ float yl[2][16][HID];
  __shared__ float zl[2][16][H2];
  int wid = threadIdx.x >> 5;
  int lane = threadIdx.x & 31;
  int tile = blockIdx.x * 2 + wid;  // grid sized exactly: no wave exits early
  int half = lane >> 4;
  int ln16 = lane & 15;
  int kb = half * 2;

  // Build y tile in LDS (coalesced over flat index)
#pragma unroll
  for (int i = 0; i < 32; ++i) {
    int flat = i * 32 + lane;
    int m = flat >> 6, c = flat & 63;
    size_t idx = (size_t)(tile * 16 + m) * HID + c;
    float sv = ssum[idx];
    yl[wid][m][c] = x[idx] + (sv > 0.f ? agg[idx] / sv : 0.f);
  }
  __syncthreads();

  // GEMM1: [16,64] @ [64,128]
  v8f acc[8];
#pragma unroll
  for (int nt = 0; nt < 8; ++nt) {
    float bb = b1[nt * 16 + ln16];
    acc[nt] = (v8f){bb, bb, bb, bb, bb, bb, bb, bb};
  }
#pragma unroll
  for (int k = 0; k < HID; k += 4) {
    v2f a;
    a[0] = yl[wid][ln16][k + kb];
    a[1] = yl[wid][ln16][k + kb + 1];
#pragma unroll
    for (int nt = 0; nt < 8; ++nt) {
      int col = nt * 16 + ln16;
      v2f b;
      b[0] = W1[(size_t)(k + kb) * H2 + col];
      b[1] = W1[(size_t)(k + kb + 1) * H2 + col];
      acc[nt] = __builtin_amdgcn_wmma_f32_16x16x4_f32(
          false, a, false, b, (short)0, acc[nt], false, false);
    }
  }

  // Register LayerNorm over the 128-wide rows + ReLU, write A-layout to LDS
#pragma unroll
  for (int r = 0; r < 8; ++r) {
    float p = 0.f;
#pragma unroll
    for (int nt = 0; nt < 8; ++nt) p += acc[nt][r];
#pragma unroll
    for (int m = 8; m >= 1; m >>= 1) p += __shfl_xor(p, m, 32);
    float mu = p * (1.f / 128.f);
    float q = 0.f;
#pragma unroll
    for (int nt = 0; nt < 8; ++nt) {
      float d = acc[nt][r] - mu;
      q += d * d;
    }
#pragma unroll
    for (int m = 8; m >= 1; m >>= 1) q += __shfl_xor(q, m, 32);
    float rstd = rsqrtf(q * (1.f / 128.f) + LN_EPS);
    int row = r + half * 8;
#pragma unroll
    for (int nt = 0; nt < 8; ++nt) {
      int col = nt * 16 + ln16;
      float v = (acc[nt][r] - mu) * rstd * g1[col] + bt1[col];
      zl[wid][row][col] = fmaxf(v, 0.f);
    }
  }
  __syncthreads();

  // GEMM2: [16,128] @ [128,64], residual add into h
  v8f acc2[4];
#pragma unroll
  for (int nt = 0; nt < 4; ++nt) {
    float bb = b2[nt * 16 + ln16];
    acc2[nt] = (v8f){bb, bb, bb, bb, bb, bb, bb, bb};
  }
#pragma unroll
  for (int k = 0; k < H2; k += 4) {
    v2f a;
    a[0] = zl[wid][ln16][k + kb];
    a[1] = zl[wid][ln16][k + kb + 1];
#pragma unroll
    for (int nt = 0; nt < 4; ++nt) {
      int col = nt * 16 + ln16;
      v2f b;
      b[0] = W2[(size_t)(k + kb) * HID + col];
      b[1] = W2[(size_t)(k + kb + 1) * HID + col];
      acc2[nt] = __builtin_amdgcn_wmma_f32_16x16x4_f32(
          false, a, false, b, (short)0, acc2[nt], false, false);
    }
  }
#pragma unroll
  for (int nt = 0; nt < 4; ++nt) {
    int col = nt * 16 + ln16;
#pragma unroll
    for (int r = 0; r < 8; ++r) {
      size_t idx = (size_t)(tile * 16 + r + half * 8) * HID + col;
      h[idx] += acc2[nt][r];
    }
  }
}

// ---------------------------------------------------------------------------
extern "C" void kernel_launch(void* const* d_in, const int* in_sizes, int n_in,
                              void* d_out, int out_size, void* d_ws,
                              size_t ws_size, hipStream_t stream) {
  const float* node_feats = (const float*)d_in[0];
  const float* edge_feats = (const float*)d_in[1];
  const int* src = (const int*)d_in[2];
  const int* dst = (const int*)d_in[3];
  const float* W_node = (const float*)d_in[4];
  const float* b_node = (const float*)d_in[5];
  const float* W_edge = (const float*)d_in[6];
  const float* b_edge = (const float*)d_in[7];
  const float* betas = (const float*)d_in[8];
  const float* W1 = (const float*)d_in[9];
  const float* b1 = (const float*)d_in[10];
  const float* ln1_g = (const float*)d_in[11];
  const float* ln1_b = (const float*)d_in[12];
  const float* W2 = (const float*)d_in[13];
  const float* b2 = (const float*)d_in[14];
  const float* norm_g = (const float*)d_in[15];
  const float* norm_b = (const float*)d_in[16];
  const float* W_out = (const float*)d_in[17];
  const float* b_out = (const float*)d_in[18];

  const size_t N64 = (size_t)N_NODES * HID;
  float* ws = (float*)d_ws;
  float* h = ws;             // [N,64]
  float* x = h + N64;        // [N,64]
  float* mx = x + N64;       // [N,64]
  float* ssum = mx + N64;    // [N,64]
  float* agg = ssum + N64;   // [N,64]
  float* eh = agg + N64;     // [E,64]

  const int node_tiles = N_NODES / 16;  // 6250
  const int edge_tiles = N_EDGES / 16;  // 75000

  // Encoders
  gemm_bias_wmma<128, 4><<<(node_tiles + 3) / 4, 128, 0, stream>>>(
      node_feats, 128, W_node, HID, b_node, h, HID, node_tiles);
  gemm_bias_wmma<8, 4><<<edge_tiles / 4, 128, 0, stream>>>(
      edge_feats, 8, W_edge, HID, b_edge, eh, HID, edge_tiles);

  const int edge_grid = (N_EDGES * 4) / 256;  // 18750
  for (int i = 0; i < LAYERS; ++i) {
    ln_relu_kernel<<<N_NODES / 8, 256, 0, stream>>>(
        h, norm_g + i * HID, norm_b + i * HID, x, mx, ssum, agg, N_NODES, 1);
    edge_pass1_kernel<<<edge_grid, 256, 0, stream>>>(x, eh, src, dst, betas, i,
                                                     mx);
    edge_pass2_kernel<<<edge_grid, 256, 0, stream>>>(x, eh, src, dst, betas, i,
                                                     mx, ssum, agg);
    mlp_fused_wmma<<<node_tiles / 2, 64, 0, stream>>>(
        x, agg, ssum, W1 + (size_t)i * HID * H2, b1 + i * H2, ln1_g + i * H2,
        ln1_b + i * H2, W2 + (size_t)i * H2 * HID, b2 + i * HID, h);
  }

  // Final norm (layer 0 params) + output projection
  ln_relu_kernel<<<N_NODES / 8, 256, 0, stream>>>(h, norm_g, norm_b, x, mx,
                                                  ssum, agg, N_NODES, 0);
  gemm_bias_wmma<64, 7><<<(node_tiles + 3) / 4, 128, 0, stream>>>(
      x, HID, W_out, OUTD, b_out, (float*)d_out, OUTD, node_tiles);
}